// NonLocalDenoiser_74440373174436
// MI455X (gfx1250) — compile-verified
//
#include <hip/hip_runtime.h>
#include <cstdint>

// ---- problem constants (from reference setup) ----
#define PATCHES   24336                    // (160-5+1)^2
#define HORF      14
#define VERF      75                       // C*PS*PS
#define IMG_H     160
#define IMG_W     160
#define NIMG      2
#define TFRAMES   (NIMG * HORF)            // 28
#define PS5       5
#define CH        3
#define NPIX      (TFRAMES * IMG_H * IMG_W)        // 716800
#define IMG_FLOATS (NPIX * CH)                     // 2150400
#define OUT_TOTAL (NIMG * PATCHES * HORF * VERF)   // 51105600

// ---- pass-1 tiling ----
#define CHUNK_J      4
#define CHUNK_ELEMS  (CHUNK_J * HORF * VERF)       // 4200 floats = 16.8 KB
#define JPB          16                            // j's per block
#define NCHUNK       (JPB / CHUNK_J)               // 4
#define BLOCKS_P1    (NIMG * (PATCHES / JPB))      // 2 * 1521 = 3042

#if defined(__AMDGCN__) && __has_builtin(__builtin_amdgcn_tensor_load_to_lds) && \
    __has_builtin(__builtin_amdgcn_s_wait_tensorcnt)
#define USE_TDM 1
#else
#define USE_TDM 0
#endif

typedef __attribute__((ext_vector_type(4))) unsigned int v4u;
typedef __attribute__((ext_vector_type(8))) int          v8i;
typedef __attribute__((ext_vector_type(4))) int          v4i;

// Native no-return f32 atomic add (global_atomic_add_f32); accumulators are
// coarse-grained device memory resident in L2, so the unsafe HW atomic is exact
// same semantics here and avoids any CAS-loop expansion.
__device__ __forceinline__ void atomic_add_f32(float* p, float v)
{
#if defined(__HIP_PLATFORM_AMD__) || defined(__AMDGCN__)
  unsafeAtomicAdd(p, v);
#else
  atomicAdd(p, v);
#endif
}

#if USE_TDM
// 1-D contiguous TDM copy: nelem f32 from global to LDS (D# per CDNA5 ISA §8).
// This toolchain's builtin takes 6 args: (g0 v4u, g1 v8i, g2 v4i, g3 v4i, g4 v8i, cpol).
__device__ __forceinline__ void tdm_load_1d(const void* gsrc, unsigned lds_byte_addr,
                                            unsigned nelem)
{
  unsigned long long ga = (unsigned long long)(uintptr_t)gsrc;
  v4u g0;
  g0.x = 1u;                                           // count=1, user mode
  g0.y = lds_byte_addr;                                // lds_addr
  g0.z = (unsigned)(ga & 0xFFFFFFFFull);               // global_addr[31:0]
  g0.w = (unsigned)((ga >> 32) & 0x01FFFFFFull)        // global_addr[56:32]
         | (2u << 30);                                 // type=2 ("image")
  v8i g1;
  g1[0] = (int)(2u << 16);                             // wg_mask=0, data_size=4B
  g1[1] = (int)((nelem & 0xFFFFu) << 16);              // tensor_dim0[15:0]
  g1[2] = (int)(((nelem >> 16) & 0xFFFFu)              // tensor_dim0[31:16]
                | (1u << 16));                         // tensor_dim1 = 1
  g1[3] = (int)((nelem & 0xFFFFu) << 16);              // tile_dim0 = nelem
  g1[4] = (int)1;                                      // tile_dim1=1, tile_dim2=0
  g1[5] = (int)nelem;                                  // tensor_dim0_stride lo
  g1[6] = 0;                                           // stride hi / dim1_stride lo
  g1[7] = 0;
  v4i z4; z4[0] = z4[1] = z4[2] = z4[3] = 0;           // 2-D tensor: groups 2/3 unused
  v8i z8; z8[0] = z8[1] = z8[2] = z8[3] = 0;
  z8[4] = z8[5] = z8[6] = z8[7] = 0;
  __builtin_amdgcn_tensor_load_to_lds(g0, g1, z4, z4, z8, 0);
}
#endif

__global__ void __launch_bounds__(256) nld_zero(float* __restrict__ p, unsigned n)
{
  unsigned i = blockIdx.x * 256u + threadIdx.x;
  if (i < n) p[i] = 0.0f;
}

// Pass 1: weighted scatter of patches into img/wimg, record per-(tt,j) base pixel.
__global__ void __launch_bounds__(256) nld_scatter(
    const float* __restrict__ x, const float* __restrict__ nlDists,
    const int* __restrict__ nlInds, float* __restrict__ img,
    float* __restrict__ wimg, int* __restrict__ bppa)
{
  __shared__ float xbuf[2][CHUNK_ELEMS];
  __shared__ float wmeta[2][CHUNK_J * HORF];
  __shared__ int   bmeta[2][CHUNK_J * HORF];

  const int tid = threadIdx.x;
  const int im  = blockIdx.x / (PATCHES / JPB);
  const int j0  = (blockIdx.x % (PATCHES / JPB)) * JPB;
  const float* xblock =
      x + ((size_t)im * PATCHES + (size_t)j0) * (HORF * VERF);  // contiguous 16*1050 f32

#if USE_TDM
  if (tid < 32) {  // wave 0 drives the TDM pipeline (EXEC ignored by tensor ops)
    tdm_load_1d(xblock, (unsigned)(uintptr_t)&xbuf[0][0], CHUNK_ELEMS);
  }
#endif

  for (int ci = 0; ci < NCHUNK; ++ci) {
    const int buf = ci & 1;
#if USE_TDM
    if (tid < 32) {
      if (ci + 1 < NCHUNK) {
        tdm_load_1d(xblock + (size_t)(ci + 1) * CHUNK_ELEMS,
                    (unsigned)(uintptr_t)&xbuf[(ci + 1) & 1][0], CHUNK_ELEMS);
        __builtin_amdgcn_s_wait_tensorcnt(1);   // chunk ci landed; ci+1 in flight
      } else {
        __builtin_amdgcn_s_wait_tensorcnt(0);
      }
    }
#endif
    // per-(tt,j) metadata for this chunk: weight + base pixel index
    const int jc0 = j0 + ci * CHUNK_J;
    if (tid < CHUNK_J * HORF) {                 // 56 entries
      const int jj = tid / HORF;
      const int hf = tid % HORF;
      const int tt = im * HORF + hf;
      const size_t row = (size_t)tt * PATCHES + (size_t)(jc0 + jj);
      const float wv = expf(-nlDists[row * 14u]);         // k = 0 only
      const int ti = nlInds[row * 42u + 0];
      const int hi = nlInds[row * 42u + 1];
      const int wi = nlInds[row * 42u + 2];
      const int base = (ti * IMG_H + hi) * IMG_W + wi;
      wmeta[buf][tid] = wv;
      bmeta[buf][tid] = base;
      bppa[row] = base;                         // reused by pass 3 (L2-resident)
    }
    __syncthreads();                            // TDM data + metadata visible

    // img scatter: 4200 native f32 atomics per chunk, all landing in L2
    for (int e = tid; e < CHUNK_ELEMS; e += 256) {
      const int jj = e / (HORF * VERF);
      const int r  = e % (HORF * VERF);
      const int hf = r / VERF;
      const int v  = r % VERF;
      const int m  = jj * HORF + hf;
      const int c  = v / (PS5 * PS5);
      const int pq = v % (PS5 * PS5);
      const int p  = pq / PS5;
      const int q  = pq % PS5;
#if USE_TDM
      const float val = xbuf[buf][e];
#else
      const float val = xblock[(size_t)ci * CHUNK_ELEMS + e];
#endif
      atomic_add_f32(&img[(size_t)(bmeta[buf][m] + p * IMG_W + q) * CH + c],
                     wmeta[buf][m] * val);
    }
    // wimg scatter: 25 weight adds per (tt,j)
    for (int e = tid; e < CHUNK_J * HORF * 25; e += 256) {
      const int m  = e / 25;
      const int pq = e % 25;
      atomic_add_f32(&wimg[bmeta[buf][m] + (pq / PS5) * IMG_W + (pq % PS5)],
                     wmeta[buf][m]);
    }
    __syncthreads();                            // protect buffers before reuse
  }
}

// Pass 2: per-pixel normalization img /= wimg.
__global__ void __launch_bounds__(256) nld_norm(float* __restrict__ img,
                                                const float* __restrict__ wimg)
{
  unsigned i = blockIdx.x * 256u + threadIdx.x;
  if (i >= NPIX) return;
  const float w = wimg[i];
  img[i * 3 + 0] /= w;
  img[i * 3 + 1] /= w;
  img[i * 3 + 2] /= w;
}

// Pass 3: gather from normalized image and emit the reference's scrambled layout
// (reshape (t,patches,verf) -> (images,hor_f,ver_f,patches) -> transpose(0,3,1,2)).
// Output writes are fully coalesced; img/bppa gathers hit L2.
__global__ void __launch_bounds__(256) nld_gather(const int* __restrict__ bppa,
                                                  const float* __restrict__ img,
                                                  float* __restrict__ out)
{
  const unsigned o = blockIdx.x * 256u + threadIdx.x;
  if (o >= OUT_TOTAL) return;
  const unsigned v2 = o % VERF;
  const unsigned r  = o / VERF;
  const unsigned hf = r % HORF;
  const unsigned r2 = r / HORF;
  const unsigned j2 = r2 % PATCHES;
  const unsigned im = r2 / PATCHES;
  const unsigned q  = v2 * PATCHES + j2;        // raw-reshape bijection
  const unsigned j  = q / VERF;
  const unsigned v  = q % VERF;
  const unsigned tt = im * HORF + hf;
  const int base = bppa[(size_t)tt * PATCHES + j];
  const unsigned c  = v / (PS5 * PS5);
  const unsigned pq = v % (PS5 * PS5);
  out[o] = img[(size_t)(base + (pq / PS5) * IMG_W + (pq % PS5)) * CH + c];
}

extern "C" void kernel_launch(void* const* d_in, const int* in_sizes, int n_in,
                              void* d_out, int out_size, void* d_ws, size_t ws_size,
                              hipStream_t stream)
{
  const float* x       = (const float*)d_in[0];
  const float* nlDists = (const float*)d_in[1];
  const int*   nlInds  = (const int*)d_in[2];
  float* out = (float*)d_out;

  // workspace layout: img (2150400 f32) | wimg (716800 f32) | bppa (681408 i32)
  float* img  = (float*)d_ws;
  float* wimg = img + IMG_FLOATS;
  int*   bppa = (int*)(wimg + NPIX);

  const unsigned nzero = IMG_FLOATS + NPIX;     // 2867200
  nld_zero<<<(nzero + 255u) / 256u, 256, 0, stream>>>(img, nzero);
  nld_scatter<<<BLOCKS_P1, 256, 0, stream>>>(x, nlDists, nlInds, img, wimg, bppa);
  nld_norm<<<(NPIX + 255u) / 256u, 256, 0, stream>>>(img, wimg);
  nld_gather<<<(OUT_TOTAL + 255u) / 256u, 256, 0, stream>>>(bppa, img, out);
}